// QSS2D_40664750358785
// MI455X (gfx1250) — compile-verified
//
#include <hip/hip_runtime.h>
#include <hip/hip_bf16.h>
#include <math.h>

// ---------------- problem constants (from reference setup_inputs) -----------
#define BATCH   4
#define HH      64
#define WW      64
#define LHW     (HH*WW)          // 4096
#define DMODEL  192
#define DIN     384              // d_inner
#define NXZ     (2*DIN)          // 768
#define NSTATE  16
#define DTR     12
#define KDIR    4
#define CPROJ   (DTR + 2*NSTATE) // 44
#define CPAD    48               // padded to 3 WMMA tiles
#define NCH     16               // scan chunks
#define LCH     (LHW/NCH)        // 256
#define SUBR    32               // u-subtile rows staged per TDM transfer
#define NSUB    (LCH/SUBR)       // 8
#define MROWS   (BATCH*LHW)      // 16384

typedef __attribute__((ext_vector_type(2))) float v2f;
typedef __attribute__((ext_vector_type(8))) float v8f;
typedef __attribute__((ext_vector_type(4))) unsigned u32x4;
typedef __attribute__((ext_vector_type(8))) int i32x8;
typedef __attribute__((ext_vector_type(4))) int i32x4;

// --------------------------- TDM availability -------------------------------
#if defined(__has_builtin)
#if __has_builtin(__builtin_amdgcn_tensor_load_to_lds)
#define HAVE_TDM 1
#endif
#endif
#ifndef HAVE_TDM
#define HAVE_TDM 0
#endif

#if HAVE_TDM
#if defined(__has_include)
#if __has_include(<hip/amd_detail/amd_gfx1250_TDM.h>)
#define TDM_6ARG 1
#endif
#endif
#ifndef TDM_6ARG
#define TDM_6ARG 0
#endif

// DMA a 2-D tile (dim0 elements/row, dim1 rows, row stride in f32 elements)
// of f32 data from global memory into LDS at byte offset ldsOff.
__device__ __forceinline__ void tdm_load2d_f32(const void* g, unsigned ldsOff,
                                               unsigned dim0, unsigned dim1,
                                               unsigned strideElems) {
  unsigned long long ga = (unsigned long long)g;
  u32x4 g0;
  g0[0] = 1u;                       // count=1, user descriptor
  g0[1] = ldsOff;                   // lds_addr
  g0[2] = (unsigned)ga;             // global_addr[31:0]
  g0[3] = (unsigned)((ga >> 32) & 0x1FFFFFFull) | (2u << 30);  // addr hi | type=2
  i32x8 g1;
  g1[0] = (int)0x00020000u;                              // data_size=4B
  g1[1] = (int)((dim0 & 0xFFFFu) << 16);                 // tensor_dim0[15:0]
  g1[2] = (int)((dim0 >> 16) | ((dim1 & 0xFFFFu) << 16));// dim0 hi | tensor_dim1 lo
  g1[3] = (int)((dim1 >> 16) | ((dim0 & 0xFFFFu) << 16));// dim1 hi | tile_dim0
  g1[4] = (int)(dim1 & 0xFFFFu);                         // tile_dim1 (tile_dim2=0)
  g1[5] = (int)strideElems;                              // tensor_dim0_stride lo32
  g1[6] = 0;
  g1[7] = 0;
  i32x4 gz; gz[0]=0; gz[1]=0; gz[2]=0; gz[3]=0;
#if TDM_6ARG
  i32x8 gz8; for (int i=0;i<8;++i) gz8[i]=0;
  __builtin_amdgcn_tensor_load_to_lds(g0, g1, gz, gz, gz8, 0);
#else
  __builtin_amdgcn_tensor_load_to_lds(g0, g1, gz, gz, 0);
#endif
}

__device__ __forceinline__ void tdm_wait0() {
#if __has_builtin(__builtin_amdgcn_s_wait_tensorcnt)
  __builtin_amdgcn_s_wait_tensorcnt(0);
#else
  asm volatile("s_wait_tensorcnt 0x0" ::: "memory");
#endif
}

__device__ __forceinline__ unsigned lds_off_u32(const void* p) {
  // generic pointer to LDS: low 32 bits are the LDS byte offset
  return (unsigned)(unsigned long long)p;
}
#endif  // HAVE_TDM

// ------------------------- WMMA helper (f32 16x16x4) ------------------------
__device__ __forceinline__ v8f wmma4(v2f a, v2f b, v8f c) {
  return __builtin_amdgcn_wmma_f32_16x16x4_f32(false, a, false, b, (short)0, c,
                                               false, false);
}
__device__ __forceinline__ v8f v8f_zero() {
  v8f r;
#pragma unroll
  for (int i = 0; i < 8; ++i) r[i] = 0.f;
  return r;
}
__device__ __forceinline__ float softplus_f(float x) {
  return (x > 20.f) ? x : __logf(1.f + __expf(x));
}
__device__ __forceinline__ float silu_f(float x) {
  return x / (1.f + __expf(-x));
}

// scan-order index -> hw-order linear index, per direction k
__device__ __forceinline__ int scan_pos(int k, int t) {
  switch (k & 3) {
    case 0:  return t;
    case 1:  return (t % HH) * WW + (t / HH);
    case 2:  return LHW - 1 - t;
    default: { int tt = LHW - 1 - t; return (tt % HH) * WW + (tt / HH); }
  }
}

// ---------------- K1/K7: fp32 WMMA GEMM, LDS-staged B panel -----------------
// block = 4 waves = 64 rows x 64 cols; B panel staged per 192-deep phase.
// LDS pair-interleaved: element (r,c) at (r>>1)*128 + c*2 + (r&1)  -> one
// aligned ds_load_b64 yields the (r, r+1) fragment pair for a column.
template <int KD, int ND>
__global__ __launch_bounds__(128) void k_gemm(const float* __restrict__ A,
                                              const float* __restrict__ Bm,
                                              float* __restrict__ C) {
  __shared__ float ldsB[96 * 128];  // 192 rows x 64 cols, pair-interleaved
  const int tid = threadIdx.x;
  const int wid = tid >> 5;
  const int lane = tid & 31;
  const int lq = lane & 15, half = lane >> 4;
  const int colGrp = blockIdx.x % (ND / 64);
  const int blkRow = blockIdx.x / (ND / 64);
  const int r0 = blkRow * 64 + wid * 16;
  const int c0 = colGrp * 64;

  v8f acc[4];
#pragma unroll
  for (int j = 0; j < 4; ++j) acc[j] = v8f_zero();

  const float* arow = A + (size_t)(r0 + lq) * KD;

  for (int k0p = 0; k0p < KD; k0p += 192) {
    __syncthreads();
    // stage B[k0p .. k0p+191][c0 .. c0+63] into LDS
    for (int idx = tid; idx < 192 * 64; idx += 128) {
      const int r = idx >> 6, c = idx & 63;
      ldsB[(r >> 1) * 128 + c * 2 + (r & 1)] =
          Bm[(size_t)(k0p + r) * ND + c0 + c];
    }
    __syncthreads();
#pragma unroll 4
    for (int kk = 0; kk < 192; kk += 4) {
      const v2f af = *(const v2f*)&arow[k0p + kk + half * 2];
      const int pp = ((kk >> 1) + half) * 128;
#pragma unroll
      for (int j = 0; j < 4; ++j) {
        const v2f bf = *(const v2f*)&ldsB[pp + (j * 16 + lq) * 2];
        acc[j] = wmma4(af, bf, acc[j]);
      }
    }
  }
#pragma unroll
  for (int j = 0; j < 4; ++j)
#pragma unroll
    for (int v = 0; v < 8; ++v)
      C[(size_t)(r0 + v + half * 8) * ND + c0 + j * 16 + lq] = acc[j][v];
}

// ---------------- K2: depthwise 3x3 conv + bias + SiLU, channels-last -------
__global__ __launch_bounds__(256) void k_conv(const float* __restrict__ XZ,
                                              const float* __restrict__ cw,
                                              const float* __restrict__ cb,
                                              float* __restrict__ XC) {
  const int idx = blockIdx.x * blockDim.x + threadIdx.x;
  if (idx >= BATCH * LHW * DIN) return;
  const int d = idx % DIN;
  const int l = (idx / DIN) % LHW;
  const int b = idx / (DIN * LHW);
  const int h = l / WW, w = l % WW;
  float s = cb[d];
#pragma unroll
  for (int kh = 0; kh < 3; ++kh) {
    const int hh = h + kh - 1;
    if ((unsigned)hh >= (unsigned)HH) continue;
#pragma unroll
    for (int kw = 0; kw < 3; ++kw) {
      const int wp = w + kw - 1;
      if ((unsigned)wp >= (unsigned)WW) continue;
      s += XZ[(size_t)(b * LHW + hh * WW + wp) * NXZ + d] *
           cw[d * 9 + kh * 3 + kw];
    }
  }
  XC[idx] = silu_f(s);
}

// ---------------- K3: x_dbl projection (WMMA, LDS-staged padded weights) ----
// block = 4 waves, all sharing one (b,k); weight 44x384 staged zero-padded
// to 48x384 in pair-interleaved LDS -> guard-free inner loop.
__global__ __launch_bounds__(128) void k_xdbl(const float* __restrict__ XC,
                                              const float* __restrict__ xpw,
                                              float* __restrict__ XDBL) {
  __shared__ float ldsW[192 * 96];  // (d-pair)*96 + c*2 + (d&1)
  const int tid = threadIdx.x;
  const int wid = tid >> 5;
  const int lane = tid & 31;
  const int lq = lane & 15, half = lane >> 4;
  const int bk = blockIdx.x / 64;          // 64 blocks per (b,k)
  const int tTile = (blockIdx.x % 64) * 4 + wid;
  const int k = bk % KDIR, b = bk / KDIR;
  const int t0 = tTile * 16;

  const float* wk = xpw + (size_t)k * CPROJ * DIN;
  for (int idx = tid; idx < CPAD * DIN; idx += 128) {
    const int d = idx % DIN, c = idx / DIN;
    ldsW[(d >> 1) * 96 + c * 2 + (d & 1)] =
        (c < CPROJ) ? wk[(size_t)c * DIN + d] : 0.f;
  }
  __syncthreads();

  const int l = scan_pos(k, t0 + lq);
  const float* arow = XC + (size_t)(b * LHW + l) * DIN;

  v8f acc[3];
#pragma unroll
  for (int j = 0; j < 3; ++j) acc[j] = v8f_zero();

#pragma unroll 4
  for (int kk = 0; kk < DIN; kk += 4) {
    const v2f af = *(const v2f*)&arow[kk + half * 2];
    const int pp = ((kk >> 1) + half) * 96;
#pragma unroll
    for (int j = 0; j < 3; ++j) {
      const v2f bf = *(const v2f*)&ldsW[pp + (j * 16 + lq) * 2];
      acc[j] = wmma4(af, bf, acc[j]);
    }
  }
  float* orow = XDBL + (size_t)bk * LHW * CPAD;
#pragma unroll
  for (int j = 0; j < 3; ++j)
#pragma unroll
    for (int v = 0; v < 8; ++v)
      orow[(size_t)(t0 + v + half * 8) * CPAD + j * 16 + lq] = acc[j][v];
}

// ---------------- K4: zero the y accumulator --------------------------------
__global__ __launch_bounds__(256) void k_zero(float* __restrict__ p, int n) {
  const int i = blockIdx.x * blockDim.x + threadIdx.x;
  if (i < n) p[i] = 0.f;
}

// -------- scan staging geometry: u sub-tile of SUBR steps as a 2-D tile -----
// All 4 directions are (row stride strL, base lbase) tiles in hw-order memory;
// reversed directions read LDS rows backwards (j = SUBR-1-r).
__device__ __forceinline__ void sub_geom(int k, int t0, int& lbase, int& strL,
                                         int& rev) {
  switch (k & 3) {
    case 0:  lbase = t0; strL = 1; rev = 0; break;
    case 1:  lbase = (t0 & 63) * WW + (t0 >> 6); strL = WW; rev = 0; break;
    case 2:  lbase = LHW - SUBR - t0; strL = 1; rev = 1; break;
    default: { const int tp = LHW - 1 - t0;
               lbase = ((tp & 63) - (SUBR - 1)) * WW + (tp >> 6);
               strL = WW; rev = 1; } break;
  }
}

// ---------------- K5a/K5c: chunked selective scan ---------------------------
// block = 384 threads (one per d); blockIdx = bk*NCH + chunk.
// xdbl slab (256x48) + u sub-tiles staged into LDS; with TDM they are DMA'd
// (double-buffered, overlapped with compute, synced by s_wait_tensorcnt).
template <int EMIT_Y>
__global__ __launch_bounds__(384) void k_scan(const float* __restrict__ XC,
                                              const float* __restrict__ XDBL,
                                              const float* __restrict__ dtw,
                                              const float* __restrict__ dtb,
                                              const float* __restrict__ A_log,
                                              const float* __restrict__ hinbuf,
                                              float* __restrict__ houtbuf,
                                              float* __restrict__ aprodbuf,
                                              float* __restrict__ YSUM) {
  __shared__ float lds_x[LCH * CPAD];       // 48 KB xdbl slab
  __shared__ float lds_u[2][SUBR * DIN];    // 2 x 48 KB u sub-tiles
  const int d = threadIdx.x;
  const int wid = threadIdx.x >> 5;
  const int chunk = blockIdx.x % NCH;
  const int bk = blockIdx.x / NCH;
  const int k = bk % KDIR, b = bk / KDIR;

  float wdt[DTR];
#pragma unroll
  for (int r = 0; r < DTR; ++r) wdt[r] = dtw[(size_t)(k * DIN + d) * DTR + r];
  const float bias = dtb[k * DIN + d];
  float Av[NSTATE];
#pragma unroll
  for (int s = 0; s < NSTATE; ++s)
    Av[s] = -__expf(A_log[(size_t)(k * DIN + d) * NSTATE + s]);

  float h[NSTATE], ap[NSTATE];
  const size_t hbase = ((size_t)(bk * NCH + chunk) * DIN + d) * NSTATE;
#pragma unroll
  for (int s = 0; s < NSTATE; ++s) {
    h[s] = EMIT_Y ? hinbuf[hbase + s] : 0.f;
    ap[s] = 1.f;
  }

  const float* slab = XDBL + (size_t)(bk * LHW + chunk * LCH) * CPAD;

#if HAVE_TDM
  {
    int lb, sl, rv; sub_geom(k, chunk * LCH, lb, sl, rv);
    if (wid == 0) {
      tdm_load2d_f32(slab, lds_off_u32(lds_x), LCH * CPAD, 1, LCH * CPAD);
      tdm_load2d_f32(XC + (size_t)(b * LHW + lb) * DIN,
                     lds_off_u32(&lds_u[0][0]), DIN, SUBR, sl * DIN);
      tdm_wait0();
    }
  }
  __syncthreads();
#else
  for (int idx = d; idx < LCH * CPAD; idx += DIN) lds_x[idx] = slab[idx];
#endif

  for (int sub = 0; sub < NSUB; ++sub) {
    const int t0 = chunk * LCH + sub * SUBR;
    int lbase, strL, rev; sub_geom(k, t0, lbase, strL, rev);
#if HAVE_TDM
    if (wid == 0 && sub + 1 < NSUB) {
      int lb, sl, rv; sub_geom(k, t0 + SUBR, lb, sl, rv);
      tdm_load2d_f32(XC + (size_t)(b * LHW + lb) * DIN,
                     lds_off_u32(&lds_u[(sub + 1) & 1][0]), DIN, SUBR,
                     sl * DIN);
    }
    const float* ubuf = &lds_u[sub & 1][0];
#else
    __syncthreads();
    for (int r = 0; r < SUBR; ++r)
      lds_u[0][r * DIN + d] = XC[(size_t)(b * LHW + lbase + r * strL) * DIN + d];
    __syncthreads();
    const float* ubuf = &lds_u[0][0];
#endif
    for (int r = 0; r < SUBR; ++r) {
      const int j = rev ? (SUBR - 1 - r) : r;   // LDS row (ascending-l order)
      const float* dr = &lds_x[(sub * SUBR + r) * CPAD];
      float xv = bias;
#pragma unroll
      for (int q = 0; q < DTR; ++q) xv += wdt[q] * dr[q];
      const float delta = softplus_f(xv);
      const float du = delta * ubuf[j * DIN + d];
      float y = 0.f;
#pragma unroll
      for (int s = 0; s < NSTATE; ++s) {
        const float e = __expf(delta * Av[s]);
        h[s] = h[s] * e + du * dr[DTR + s];
        if (EMIT_Y) y += h[s] * dr[DTR + NSTATE + s];
        else        ap[s] *= e;
      }
      if (EMIT_Y) {
        const int l = lbase + j * strL;
        atomicAdd(&YSUM[(size_t)(b * LHW + l) * DIN + d], y);
      }
    }
#if HAVE_TDM
    if (wid == 0) tdm_wait0();
    __syncthreads();
#endif
  }

  if (!EMIT_Y) {
#pragma unroll
    for (int s = 0; s < NSTATE; ++s) {
      houtbuf[hbase + s] = h[s];
      aprodbuf[hbase + s] = ap[s];
    }
  }
}

// ---------------- K5b: stitch chunk summaries -------------------------------
__global__ __launch_bounds__(384) void k_scan2(const float* __restrict__ houtbuf,
                                               const float* __restrict__ aprodbuf,
                                               float* __restrict__ hinbuf) {
  const int d = threadIdx.x;
  const int bk = blockIdx.x;
#pragma unroll
  for (int s = 0; s < NSTATE; ++s) {
    float hin = 0.f;
    for (int c = 0; c < NCH; ++c) {
      const size_t base = ((size_t)(bk * NCH + c) * DIN + d) * NSTATE + s;
      hinbuf[base] = hin;
      hin = aprodbuf[base] * hin + houtbuf[base];
    }
  }
}

// ---------------- K6: +D*u, LayerNorm over d_inner, gate with silu(z) -------
__global__ __launch_bounds__(384) void k_norm(const float* __restrict__ YSUM,
                                              const float* __restrict__ XZ,
                                              const float* __restrict__ Ds,
                                              const float* __restrict__ ln_g,
                                              const float* __restrict__ ln_b,
                                              float* __restrict__ XC) {
  const int d = threadIdx.x;
  const int bl = blockIdx.x;
  const float sumD = Ds[0 * DIN + d] + Ds[1 * DIN + d] +
                     Ds[2 * DIN + d] + Ds[3 * DIN + d];
  const float u = XC[(size_t)bl * DIN + d];
  const float y = YSUM[(size_t)bl * DIN + d] + sumD * u;

  float v1 = y, v2 = y * y;
#pragma unroll
  for (int off = 16; off > 0; off >>= 1) {
    v1 += __shfl_xor(v1, off, 32);
    v2 += __shfl_xor(v2, off, 32);
  }
  __shared__ float s1[DIN / 32], s2[DIN / 32];
  const int wid = threadIdx.x >> 5;
  if ((threadIdx.x & 31) == 0) { s1[wid] = v1; s2[wid] = v2; }
  __syncthreads();
  if (threadIdx.x == 0) {
    float a = 0.f, bb = 0.f;
#pragma unroll
    for (int i = 0; i < DIN / 32; ++i) { a += s1[i]; bb += s2[i]; }
    s1[0] = a; s2[0] = bb;
  }
  __syncthreads();
  const float mean = s1[0] * (1.f / DIN);
  const float var = s2[0] * (1.f / DIN) - mean * mean;
  const float yn = (y - mean) * rsqrtf(var + 1e-5f) * ln_g[d] + ln_b[d];
  const float z = XZ[(size_t)bl * NXZ + DIN + d];
  XC[(size_t)bl * DIN + d] = yn * silu_f(z);
}

// ---------------------------------------------------------------------------
extern "C" void kernel_launch(void* const* d_in, const int* in_sizes, int n_in,
                              void* d_out, int out_size, void* d_ws,
                              size_t ws_size, hipStream_t stream) {
  const float* x      = (const float*)d_in[0];
  const float* W_in   = (const float*)d_in[1];
  const float* conv_w = (const float*)d_in[2];
  const float* conv_b = (const float*)d_in[3];
  const float* xpw    = (const float*)d_in[4];
  const float* dtw    = (const float*)d_in[5];
  const float* dtb    = (const float*)d_in[6];
  const float* A_log  = (const float*)d_in[7];
  const float* Ds     = (const float*)d_in[8];
  const float* ln_g   = (const float*)d_in[9];
  const float* ln_b   = (const float*)d_in[10];
  const float* W_out  = (const float*)d_in[11];
  float* out = (float*)d_out;

  float* ws = (float*)d_ws;
  float* xz    = ws;
  float* xc    = xz    + (size_t)MROWS * NXZ;
  float* xdbl  = xc    + (size_t)MROWS * DIN;
  float* hout  = xdbl  + (size_t)BATCH * KDIR * LHW * CPAD;
  float* aprod = hout  + (size_t)BATCH * KDIR * NCH * DIN * NSTATE;
  float* hin   = aprod + (size_t)BATCH * KDIR * NCH * DIN * NSTATE;
  float* ysum  = hin   + (size_t)BATCH * KDIR * NCH * DIN * NSTATE;

  // 1) xz = x @ W_in  (fp32 WMMA, LDS-staged B panel)
  k_gemm<DMODEL, NXZ><<<(MROWS / 64) * (NXZ / 64), 128, 0, stream>>>(x, W_in,
                                                                     xz);
  // 2) depthwise conv + SiLU -> xc (channels-last)
  {
    const int n = BATCH * LHW * DIN;
    k_conv<<<(n + 255) / 256, 256, 0, stream>>>(xz, conv_w, conv_b, xc);
  }
  // 3) x_dbl projection (fp32 WMMA, guard-free via LDS zero-padded weights)
  k_xdbl<<<BATCH * KDIR * 64, 128, 0, stream>>>(xc, xpw, xdbl);
  // 4) zero y accumulator
  {
    const int n = MROWS * DIN;
    k_zero<<<(n + 255) / 256, 256, 0, stream>>>(ysum, n);
  }
  // 5) chunked selective scan (TDM-staged LDS when available)
  k_scan<0><<<BATCH * KDIR * NCH, DIN, 0, stream>>>(xc, xdbl, dtw, dtb, A_log,
                                                    nullptr, hout, aprod,
                                                    nullptr);
  k_scan2<<<BATCH * KDIR, DIN, 0, stream>>>(hout, aprod, hin);
  k_scan<1><<<BATCH * KDIR * NCH, DIN, 0, stream>>>(xc, xdbl, dtw, dtb, A_log,
                                                    hin, nullptr, nullptr,
                                                    ysum);
  // 6) +D*u, LayerNorm, silu(z) gate (in-place into xc)
  k_norm<<<MROWS, DIN, 0, stream>>>(ysum, xz, Ds, ln_g, ln_b, xc);
  // 7) out = g @ W_out (fp32 WMMA)
  k_gemm<DIN, DMODEL><<<(MROWS / 64) * (DMODEL / 64), 128, 0, stream>>>(xc,
                                                                        W_out,
                                                                        out);
}